// Frankenstein_28784870818475
// MI455X (gfx1250) — compile-verified
//
#include <hip/hip_runtime.h>
#include <math.h>

#define XN 47764
#define HN 128
#define LN 8
#define VTN 3620
#define WN 32
#define RN 8
#define NN 512
#define DN 48276           // X + R*W + 2*H
#define ETN 395
#define PN 48148           // X + W*R + H  (parallel prefix incl. hidden_prev)
#define CHUNK 1024
#define NCHUNK 48          // ceil(PN / CHUNK)
#define EPSF 1e-8f

typedef float v2f __attribute__((ext_vector_type(2)));
typedef float v8f __attribute__((ext_vector_type(8)));

__device__ __forceinline__ float sigf(float x) { return 1.0f / (1.0f + expf(-x)); }
__device__ __forceinline__ float softplusf(float z) {
  return fmaxf(z, 0.0f) + log1pf(expf(-fabsf(z)));
}

// Branch-free WMMA GEMV step: distribute x[d..d+3] (held across lanes as xi)
// into A row M=0 (lanes 0/16) via ds_bpermute, everything else zero.
__device__ __forceinline__ v8f gemv_step(v8f acc, int xi, int i, int base_s,
                                         bool isrow0, const float* wb) {
  int a0i = __builtin_amdgcn_ds_bpermute((i + base_s) << 2, xi);
  int a1i = __builtin_amdgcn_ds_bpermute((i + base_s + 1) << 2, xi);
  v2f a;
  a[0] = isrow0 ? __int_as_float(a0i) : 0.0f;
  a[1] = isrow0 ? __int_as_float(a1i) : 0.0f;
  v2f bv;
  bv[0] = wb[0];
  bv[1] = wb[HN];
  return __builtin_amdgcn_wmma_f32_16x16x4_f32(false, a, false, bv, (short)0, acc, false, false);
}

// ---------------------------------------------------------------- A0: build per-layer input row
__global__ void k_xcat(const float* __restrict__ x_in, const float* __restrict__ lrv,
                       const float* __restrict__ hprev, float* __restrict__ xcat) {
  int tid = blockIdx.x * 256 + threadIdx.x;
  if (tid >= LN * PN) return;
  int l = tid / PN, d = tid % PN;
  float v;
  if (d < XN) v = x_in[d];
  else if (d < XN + WN * RN) v = lrv[d - XN];
  else v = hprev[l * HN + (d - (XN + WN * RN))];
  xcat[tid] = v;
}

// ---------------------------------------------------------------- A1: big WMMA GEMV (791 MB stream)
__global__ void __launch_bounds__(256) k_gemv_big(
    const float* __restrict__ Wi, const float* __restrict__ Wf,
    const float* __restrict__ Wo, const float* __restrict__ Ws,
    const float* __restrict__ xcat, float* __restrict__ partials) {
  int b = blockIdx.x;                 // 1536 = 48 chunks * 8 layers * 4 gates
  int c = b >> 5;
  int l = (b >> 2) & 7;
  int g = b & 3;
  int wave = threadIdx.x >> 5;        // h-tile 0..7
  int lane = threadIdx.x & 31;
  const float* Wg = (g == 0) ? Wi : (g == 1) ? Wf : (g == 2) ? Wo : Ws;
  int h0 = wave << 4;
  int d0 = c * CHUNK;
  int dend = (d0 + CHUNK < PN) ? (d0 + CHUNK) : PN;
  const float* xrow = xcat + (size_t)l * PN;
  // B layout: lanes 0-15 -> rows d,d+1 ; lanes 16-31 -> rows d+2,d+3 ; col = h0 + (lane&15)
  const float* wb = Wg + ((size_t)l * DN + d0) * HN + ((lane >> 4) << 8) + h0 + (lane & 15);
  int base_s = (lane >> 4) << 1;      // 0 or 2
  bool isrow0 = ((lane & 15) == 0);
  v8f acc = {};
  for (int dblk = d0; dblk < dend; dblk += 32) {
    int xi = __float_as_int(xrow[dblk + lane]);   // 32 x-values, one coalesced load
    int nds = dend - dblk;
    if (nds >= 32) {
#pragma unroll
      for (int i = 0; i < 32; i += 4) {
        acc = gemv_step(acc, xi, i, base_s, isrow0, wb);
        wb += 4 * HN;
      }
    } else {
      for (int i = 0; i < nds; i += 4) {
        acc = gemv_step(acc, xi, i, base_s, isrow0, wb);
        wb += 4 * HN;
      }
    }
  }
  if (lane < 16)
    partials[((size_t)c * 32 + l * 4 + g) * HN + h0 + lane] = acc[0];
}

// ---------------------------------------------------------------- B: deterministic chunk reduce + bias
__global__ void k_reduce_pre(const float* __restrict__ partials,
                             const float* __restrict__ b_i, const float* __restrict__ b_f,
                             const float* __restrict__ b_o, const float* __restrict__ b_s,
                             float* __restrict__ pre) {
  int tid = blockIdx.x * 256 + threadIdx.x;
  if (tid >= LN * 4 * HN) return;
  int h = tid & 127, g = (tid >> 7) & 3, l = tid >> 9;
  const float* bb = (g == 0) ? b_i : (g == 1) ? b_f : (g == 2) ? b_o : b_s;
  float s = bb[l * HN + h];
  const float* p = partials + (size_t)(l * 4 + g) * HN + h;
  for (int c = 0; c < NCHUNK; ++c) s += p[(size_t)c * 32 * HN];
  pre[(l * 4 + g) * HN + h] = s;
}

// ---------------------------------------------------------------- C: sequential recurrent tail
// 512 threads: thread = g*128 + h ; each gate group streams its own weights.
__global__ void __launch_bounds__(512) k_lstm(const float* __restrict__ pre,
    const float* __restrict__ Wi, const float* __restrict__ Wf,
    const float* __restrict__ Wo, const float* __restrict__ Ws,
    const float* __restrict__ old_states, float* __restrict__ flat) {
  __shared__ float sh[HN];
  __shared__ float sg[4][HN];
  int t = threadIdx.x;
  int h = t & 127;
  int g = t >> 7;                                   // 0:i 1:f 2:o 3:s
  const float* Wg = (g == 0) ? Wi : (g == 1) ? Wf : (g == 2) ? Wo : Ws;
  if (t < HN) sh[t] = 0.0f;
  __syncthreads();
  for (int l = 0; l < LN; ++l) {
    float acc = pre[(l * 4 + g) * HN + h];
    size_t base = ((size_t)l * DN + PN) * HN + h;   // h_prev_layer rows start at d = PN
#pragma unroll 4
    for (int j = 0; j < HN; ++j)
      acc += sh[j] * Wg[base + (size_t)j * HN];
    sg[g][h] = acc;
    __syncthreads();
    if (g == 0) {
      float ig = sigf(sg[0][h]);
      float fg = sigf(sg[1][h]);
      float og = sigf(sg[2][h]);
      float ss = tanhf(sg[3][h]);
      float st = fg * old_states[l * HN + h] + ig * ss;
      float hn = og * tanhf(st);
      sh[h] = hn;
      flat[l * HN + h] = hn;
    }
    __syncthreads();
  }
}

// ---------------------------------------------------------------- D: WMMA GEMV for W_y and W_E
__global__ void __launch_bounds__(256) k_gemv_proj(
    const float* __restrict__ Wy, const float* __restrict__ WE,
    const float* __restrict__ flat, float* __restrict__ outy, float* __restrict__ outE) {
  int t = blockIdx.x * 8 + (threadIdx.x >> 5);      // column tile
  int lane = threadIdx.x & 31;
  const float* M; float* O; int NC, col0;
  if (t < 227)      { M = Wy; O = outy; NC = VTN; col0 = t * 16; }
  else if (t < 252) { M = WE; O = outE; NC = ETN; col0 = (t - 227) * 16; }
  else return;
  int col = col0 + (lane & 15);
  int colc = (col < NC) ? col : (NC - 1);           // clamp for partial tiles
  const float* wb = M + (size_t)((lane >> 4) ? 2 * NC : 0) + colc;
  int base_s = (lane >> 4) << 1;
  bool isrow0 = ((lane & 15) == 0);
  v8f acc = {};
  for (int dblk = 0; dblk < LN * HN; dblk += 32) {  // 1024 = 32 * 32
    int xi = __float_as_int(flat[dblk + lane]);
#pragma unroll
    for (int i = 0; i < 32; i += 4) {
      int a0i = __builtin_amdgcn_ds_bpermute((i + base_s) << 2, xi);
      int a1i = __builtin_amdgcn_ds_bpermute((i + base_s + 1) << 2, xi);
      v2f a;
      a[0] = isrow0 ? __int_as_float(a0i) : 0.0f;
      a[1] = isrow0 ? __int_as_float(a1i) : 0.0f;
      v2f bv;
      bv[0] = wb[0];
      bv[1] = wb[NC];
      acc = __builtin_amdgcn_wmma_f32_16x16x4_f32(false, a, false, bv, (short)0, acc, false, false);
      wb += (size_t)4 * NC;
    }
  }
  if (lane < 16 && col < NC) O[col] = acc[0];
}

// ---------------------------------------------------------------- E1: write head (usage/sort/alloc/cw/ww/memory_new)
__global__ void __launch_bounds__(512) k_write_head(
    const float* __restrict__ Ev, const float* __restrict__ mem,
    const float* __restrict__ lrw, const float* __restrict__ lu,
    const float* __restrict__ lww, float* __restrict__ ww_out, float* __restrict__ memn) {
  __shared__ float s_fg[RN], s_wkey[WN], s_erase[WN], s_wvec[WN];
  __shared__ float s_ag, s_wg, s_wstr, s_knorm;
  __shared__ float s_val[NN];
  __shared__ int   s_idx[NN];
  __shared__ float s_red[NN];
  __shared__ float s_alloc[NN];
  int t = threadIdx.x;
  if (t < RN) s_fg[t] = sigf(Ev[361 + t]);
  if (t < WN) { s_wkey[t] = Ev[264 + t]; s_erase[t] = sigf(Ev[297 + t]); s_wvec[t] = Ev[329 + t]; }
  if (t == 0) {
    s_ag = sigf(Ev[369]); s_wg = sigf(Ev[370]);
    s_wstr = 1.0f + softplusf(-Ev[296]);
    float ss = 0.0f;
    for (int w = 0; w < WN; ++w) { float k = Ev[264 + w]; ss += k * k; }
    s_knorm = fmaxf(sqrtf(ss), EPSF);
  }
  __syncthreads();
  // retention & usage
  float psi = 1.0f;
  for (int r = 0; r < RN; ++r) psi *= (1.0f - s_fg[r] * lrw[t * RN + r]);
  float u = (lu[t] + lww[t] - lu[t] * lww[t]) * psi;
  s_val[t] = u; s_idx[t] = t;
  __syncthreads();
  // stable ascending bitonic sort on (value, index)
  for (int k = 2; k <= NN; k <<= 1) {
    for (int j = k >> 1; j > 0; j >>= 1) {
      int ixj = t ^ j;
      if (ixj > t) {
        float v1 = s_val[t], v2 = s_val[ixj];
        int i1 = s_idx[t], i2 = s_idx[ixj];
        bool gt = (v1 > v2) || (v1 == v2 && i1 > i2);
        bool up = ((t & k) == 0);
        if (gt == up) {
          s_val[t] = v2; s_val[ixj] = v1;
          s_idx[t] = i2; s_idx[ixj] = i1;
        }
      }
      __syncthreads();
    }
  }
  // sequential exclusive cumprod -> allocation weighting (scattered to original slots)
  if (t == 0) {
    float run = 1.0f;
    for (int k = 0; k < NN; ++k) {
      s_alloc[s_idx[k]] = (1.0f - s_val[k]) * run;
      run *= s_val[k];
    }
  }
  __syncthreads();
  // cosine content weighting on old memory
  const float* row = mem + t * WN;
  float ss = 0.0f;
  for (int w = 0; w < WN; ++w) ss += row[w] * row[w];
  float rn = fmaxf(sqrtf(ss), EPSF);
  float dot = 0.0f;
  for (int w = 0; w < WN; ++w) dot += (row[w] / rn) * (s_wkey[w] / s_knorm);
  float score = s_wstr * dot;
  s_red[t] = score; __syncthreads();
  for (int st = NN / 2; st > 0; st >>= 1) { if (t < st) s_red[t] = fmaxf(s_red[t], s_red[t + st]); __syncthreads(); }
  float m = s_red[0]; __syncthreads();
  float e = expf(score - m);
  s_red[t] = e; __syncthreads();
  for (int st = NN / 2; st > 0; st >>= 1) { if (t < st) s_red[t] += s_red[t + st]; __syncthreads(); }
  float cw = e / s_red[0];
  float ww = s_wg * (s_ag * s_alloc[t] + (1.0f - s_ag) * cw);
  ww_out[t] = ww;
  float* mo = memn + t * WN;
  for (int w = 0; w < WN; ++w)
    mo[w] = row[w] * (1.0f - ww * s_erase[w]) + ww * s_wvec[w];
}

// ---------------------------------------------------------------- E2: temporal linkage fw/bw (Lm never materialized)
__global__ void __launch_bounds__(128) k_linkage(const float* __restrict__ ww,
    const float* __restrict__ prec, const float* __restrict__ link,
    const float* __restrict__ lrw, float* __restrict__ fw, float* __restrict__ bw) {
  __shared__ float s[128 * 16];
  int i = blockIdx.x;
  int t = threadIdx.x;
  float wwi = ww[i];
  float pi = prec[i];
  float fa[RN] = {0}, ba[RN] = {0};
  for (int j = t; j < NN; j += 128) {
    if (j == i) continue;                          // zero diagonal
    float wwj = ww[j];
    float lij = link[(size_t)i * NN + j];
    float lji = link[(size_t)j * NN + i];
    float lmf = (1.0f - wwi - wwj) * lij + wwi * prec[j];
    float lmb = (1.0f - wwj - wwi) * lji + wwj * pi;
    for (int r = 0; r < RN; ++r) {
      float lr = lrw[j * RN + r];
      fa[r] += lmf * lr;
      ba[r] += lmb * lr;
    }
  }
  for (int r = 0; r < RN; ++r) { s[t * 16 + r] = fa[r]; s[t * 16 + 8 + r] = ba[r]; }
  __syncthreads();
  for (int st = 64; st > 0; st >>= 1) {
    if (t < st) for (int k = 0; k < 16; ++k) s[t * 16 + k] += s[(t + st) * 16 + k];
    __syncthreads();
  }
  if (t < RN) fw[i * RN + t] = s[t];
  else if (t < 16) bw[i * RN + (t - 8)] = s[t];
}

// ---------------------------------------------------------------- E3: read head -> rv
__global__ void __launch_bounds__(512) k_read(const float* __restrict__ Ev,
    const float* __restrict__ memn, const float* __restrict__ fw,
    const float* __restrict__ bw, float* __restrict__ rv) {
  __shared__ float s_rs[RN], s_rm[RN * 3], s_knr[RN];
  __shared__ float s_red[NN];
  __shared__ float s_rw[NN * RN];
  int t = threadIdx.x;
  if (t < RN) {
    s_rs[t] = 1.0f + softplusf(-Ev[256 + t]);
    float a0 = Ev[371 + t * 3 + 0], a1 = Ev[371 + t * 3 + 1], a2 = Ev[371 + t * 3 + 2];
    float mm = fmaxf(a0, fmaxf(a1, a2));
    float e0 = expf(a0 - mm), e1 = expf(a1 - mm), e2 = expf(a2 - mm);
    float sm = e0 + e1 + e2;
    s_rm[t * 3 + 0] = e0 / sm; s_rm[t * 3 + 1] = e1 / sm; s_rm[t * 3 + 2] = e2 / sm;
    float ss = 0.0f;
    for (int w = 0; w < WN; ++w) { float k = Ev[w * RN + t]; ss += k * k; }
    s_knr[t] = fmaxf(sqrtf(ss), EPSF);
  }
  __syncthreads();
  const float* row = memn + t * WN;
  float ss = 0.0f;
  for (int w = 0; w < WN; ++w) ss += row[w] * row[w];
  float rn = fmaxf(sqrtf(ss), EPSF);
  float score[RN];
  for (int r = 0; r < RN; ++r) {
    float dot = 0.0f;
    for (int w = 0; w < WN; ++w) dot += (row[w] / rn) * (Ev[w * RN + r] / s_knr[r]);
    score[r] = s_rs[r] * dot;
  }
  float cr[RN];
  for (int r = 0; r < RN; ++r) {
    s_red[t] = score[r]; __syncthreads();
    for (int st = NN / 2; st > 0; st >>= 1) { if (t < st) s_red[t] = fmaxf(s_red[t], s_red[t + st]); __syncthreads(); }
    float m = s_red[0]; __syncthreads();
    float e = expf(score[r] - m);
    s_red[t] = e; __syncthreads();
    for (int st = NN / 2; st > 0; st >>= 1) { if (t < st) s_red[t] += s_red[t + st]; __syncthreads(); }
    cr[r] = e / s_red[0];
    __syncthreads();
  }
  for (int r = 0; r < RN; ++r)
    s_rw[t * RN + r] = bw[t * RN + r] * s_rm[r * 3 + 0] + cr[r] * s_rm[r * 3 + 1] + fw[t * RN + r] * s_rm[r * 3 + 2];
  __syncthreads();
  if (t < WN * RN) {
    int w = t >> 3, r = t & 7;
    float acc = 0.0f;
    for (int n = 0; n < NN; ++n) acc += memn[n * WN + w] * s_rw[n * RN + r];
    rv[t] = acc;                                   // flat index w*R + r
  }
}

// ---------------------------------------------------------------- E4: y = output + rv @ W_r
__global__ void __launch_bounds__(256) k_final(const float* __restrict__ rv,
    const float* __restrict__ Wr, const float* __restrict__ outy, float* __restrict__ y) {
  __shared__ float s_rv[WN * RN];
  int t = threadIdx.x;
  s_rv[t] = rv[t];                                 // 256 threads == W*R
  __syncthreads();
  int v = blockIdx.x * 256 + t;
  if (v >= VTN) return;
  float acc = outy[v];
  for (int k = 0; k < WN * RN; ++k) acc += s_rv[k] * Wr[(size_t)k * VTN + v];
  y[v] = acc;
}

extern "C" void kernel_launch(void* const* d_in, const int* in_sizes, int n_in,
                              void* d_out, int out_size, void* d_ws, size_t ws_size,
                              hipStream_t stream) {
  (void)in_sizes; (void)n_in; (void)out_size; (void)ws_size;
  const float* x_in   = (const float*)d_in[0];
  const float* W_i    = (const float*)d_in[1];
  const float* b_i    = (const float*)d_in[2];
  const float* W_f    = (const float*)d_in[3];
  const float* b_f    = (const float*)d_in[4];
  const float* W_o    = (const float*)d_in[5];
  const float* b_o    = (const float*)d_in[6];
  const float* W_s    = (const float*)d_in[7];
  const float* b_s    = (const float*)d_in[8];
  const float* W_y    = (const float*)d_in[9];
  const float* W_E    = (const float*)d_in[10];
  const float* W_r    = (const float*)d_in[11];
  const float* memory = (const float*)d_in[12];
  const float* lrv    = (const float*)d_in[13];
  const float* hprev  = (const float*)d_in[14];
  const float* ostate = (const float*)d_in[15];
  const float* prec   = (const float*)d_in[16];
  const float* link   = (const float*)d_in[17];
  const float* lrw    = (const float*)d_in[18];
  const float* lu     = (const float*)d_in[19];
  const float* lww    = (const float*)d_in[20];
  float* ws = (float*)d_ws;
  float* y  = (float*)d_out;

  // workspace layout (floats)
  const size_t o_xcat = 0;                                 // L*PN = 385184
  const size_t o_part = o_xcat + (size_t)LN * PN;          // 48*32*128 = 196608
  const size_t o_pre  = o_part + (size_t)NCHUNK * 32 * HN; // 4096
  const size_t o_flat = o_pre  + (size_t)LN * 4 * HN;      // 1024
  const size_t o_out  = o_flat + (size_t)LN * HN;          // 3620
  const size_t o_E    = o_out  + VTN;                      // 395
  const size_t o_ww   = o_E    + 512;                      // 512
  const size_t o_memn = o_ww   + NN;                       // 16384
  const size_t o_fw   = o_memn + (size_t)NN * WN;          // 4096
  const size_t o_bw   = o_fw   + (size_t)NN * RN;          // 4096
  const size_t o_rv   = o_bw   + (size_t)NN * RN;          // 256

  k_xcat<<<(LN * PN + 255) / 256, 256, 0, stream>>>(x_in, lrv, hprev, ws + o_xcat);
  k_gemv_big<<<NCHUNK * 32, 256, 0, stream>>>(W_i, W_f, W_o, W_s, ws + o_xcat, ws + o_part);
  k_reduce_pre<<<16, 256, 0, stream>>>(ws + o_part, b_i, b_f, b_o, b_s, ws + o_pre);
  k_lstm<<<1, 512, 0, stream>>>(ws + o_pre, W_i, W_f, W_o, W_s, ostate, ws + o_flat);
  k_gemv_proj<<<32, 256, 0, stream>>>(W_y, W_E, ws + o_flat, ws + o_out, ws + o_E);
  k_write_head<<<1, 512, 0, stream>>>(ws + o_E, memory, lrw, lu, lww, ws + o_ww, ws + o_memn);
  k_linkage<<<NN, 128, 0, stream>>>(ws + o_ww, prec, link, lrw, ws + o_fw, ws + o_bw);
  k_read<<<1, 512, 0, stream>>>(ws + o_E, ws + o_memn, ws + o_fw, ws + o_bw, ws + o_rv);
  k_final<<<(VTN + 255) / 256, 256, 0, stream>>>(ws + o_rv, W_r, ws + o_out, y);
}